// GCNNet_42563125903764
// MI455X (gfx1250) — compile-verified
//
#include <hip/hip_runtime.h>
#include <hip/hip_bf16.h>
#include <math.h>

typedef float v2f __attribute__((ext_vector_type(2)));
typedef float v8f __attribute__((ext_vector_type(8)));

// ---------------------------------------------------------------------------
// WMMA fp32 GEMM: C[M,N] = A[M,K] @ B[K,N]   (row-major, N <= NTILES*16)
// Requirements used by this launcher: K % KC == 0, blockDim = 256 (8 wave32).
// Each block computes a 128-row strip; each wave owns a 16-row strip and all
// NTILES column tiles (accumulators held in registers across the K loop).
// B is staged through LDS in KC-row chunks (padded stride -> no bank
// conflicts); A fragments for a whole chunk are prefetched into registers
// before the fully-unrolled WMMA burst so global latency is batched.
// Fragment layouts per CDNA5 ISA 7.12.2:
//   A 16x4 f32 : lane l holds A[m= l&15][k = (l>>4)*2 + {0,1}]
//   B  4x16 f32: lane l holds B[k = (l>>4)*2 + {0,1}][n = l&15]
//   C/D 16x16  : lane l, vgpr i holds C[(l>>4)*8 + i][l&15]
// ---------------------------------------------------------------------------
template <int NTILES>
__global__ void wmma_gemm_kernel(const float* __restrict__ A,
                                 const float* __restrict__ B,
                                 float* __restrict__ C,
                                 int M, int N, int K) {
    constexpr int S   = NTILES * 16 + 8;  // padded LDS row stride (floats)
    constexpr int KC  = 64;               // K-chunk rows staged in LDS
    constexpr int NKF = KC / 4;           // A fragments per chunk (16)
    extern __shared__ float sB[];         // KC * S floats

    const int tid  = threadIdx.x;
    const int wave = tid >> 5;
    const int lane = tid & 31;
    const int m0   = blockIdx.x * 128 + wave * 16;  // wave's output row base
    const int half = lane >> 4;                     // 0: lanes 0-15, 1: 16-31
    const int l16  = lane & 15;
    const int kb   = half * 2;                      // K sub-offset in fragment
    const int mr   = m0 + l16;                      // A row for this lane

    v8f acc[NTILES];
    const v8f vzero = {0.f, 0.f, 0.f, 0.f, 0.f, 0.f, 0.f, 0.f};
#pragma unroll
    for (int t = 0; t < NTILES; ++t) acc[t] = vzero;

    const bool active = (m0 < M);
    const float* Arow = A + (size_t)(active ? mr : 0) * K + kb;

    for (int kc = 0; kc < K; kc += KC) {
        __syncthreads();  // protect LDS from previous chunk's readers
        // cooperative stage of B rows [kc, kc+KC) into LDS, zero-padded cols.
        // S is constexpr so idx/S strength-reduces (no real division).
        for (int idx = tid; idx < KC * S; idx += 256) {
            const int k = idx / S, n = idx - k * S;
            sB[idx] = (n < N) ? B[(size_t)(kc + k) * N + n] : 0.0f;
        }
        __syncthreads();

        if (active) {
            // batch-prefetch all A fragments of this chunk (16x b64 loads)
            v2f af[NKF];
#pragma unroll
            for (int j = 0; j < NKF; ++j) {
                af[j].x = Arow[kc + 4 * j];
                af[j].y = Arow[kc + 4 * j + 1];
            }
            // fully-unrolled WMMA burst: registers + LDS only
#pragma unroll
            for (int j = 0; j < NKF; ++j) {
                const float* b0 = sB + (4 * j + kb) * S + l16;
#pragma unroll
                for (int t = 0; t < NTILES; ++t) {
                    v2f b;
                    b.x = b0[t * 16];
                    b.y = b0[t * 16 + S];
                    acc[t] = __builtin_amdgcn_wmma_f32_16x16x4_f32(
                        false, af[j], false, b, (short)0, acc[t], false, false);
                }
            }
        }
    }

    if (active) {
        const int rbase = m0 + half * 8;
#pragma unroll
        for (int t = 0; t < NTILES; ++t) {
            const int colI = t * 16 + l16;
            if (colI < N) {
#pragma unroll
                for (int i = 0; i < 8; ++i) {
                    const int r = rbase + i;
                    if (r < M) C[(size_t)r * N + colI] = acc[t][i];
                }
            }
        }
    }
}

// ---------------------------------------------------------------------------
// Degree / normalization
// ---------------------------------------------------------------------------
__global__ void deg_init_kernel(float* __restrict__ deg, int N) {
    int i = blockIdx.x * blockDim.x + threadIdx.x;
    if (i < N) deg[i] = 1.0f;  // self-loop
}

__global__ void deg_edges_kernel(const int* __restrict__ col,
                                 float* __restrict__ deg, int E) {
    int e = blockIdx.x * blockDim.x + threadIdx.x;
    if (e < E) unsafeAtomicAdd(&deg[col[e]], 1.0f);
}

__global__ void rsqrt_kernel(float* __restrict__ deg, int N) {
    int i = blockIdx.x * blockDim.x + threadIdx.x;
    if (i < N) {
        float d = deg[i];
        deg[i] = (d > 0.0f) ? rsqrtf(d) : 0.0f;  // now holds deg^{-1/2}
    }
}

// h[i,f] = xw[i,f] * dis[i]^2   (self-loop message; also zero-initializes h)
__global__ void selfloop_init_kernel(const float* __restrict__ xw,
                                     const float* __restrict__ dis,
                                     float* __restrict__ h, int total, int F) {
    int i = blockIdx.x * blockDim.x + threadIdx.x;
    if (i < total) {
        int node = i / F;
        float d = dis[node];
        h[i] = xw[i] * d * d;
    }
}

// scatter-add: h[col[e], :] += xw[row[e], :] * dis[row]*dis[col]
// one thread per (edge, 4-feature group); float4 gather, 4x global f32 atomics
__global__ void edge_scatter_kernel(const float* __restrict__ xw,
                                    const float* __restrict__ dis,
                                    const int* __restrict__ row,
                                    const int* __restrict__ col,
                                    float* __restrict__ h,
                                    int E, int F4, int F) {
    int tid = blockIdx.x * blockDim.x + threadIdx.x;
    if (tid >= E * F4) return;
    int e = tid / F4;
    int q = tid - e * F4;
    int r = row[e], c = col[e];
    float nrm = dis[r] * dis[c];
    const float4 v = ((const float4*)(xw + (size_t)r * F))[q];
    float* dst = h + (size_t)c * F + q * 4;
    unsafeAtomicAdd(dst + 0, v.x * nrm);
    unsafeAtomicAdd(dst + 1, v.y * nrm);
    unsafeAtomicAdd(dst + 2, v.z * nrm);
    unsafeAtomicAdd(dst + 3, v.w * nrm);
}

__global__ void bias_relu_kernel(float* __restrict__ h,
                                 const float* __restrict__ b,
                                 int total, int F) {
    int i = blockIdx.x * blockDim.x + threadIdx.x;
    if (i < total) {
        float v = h[i] + b[i % F];
        h[i] = v > 0.0f ? v : 0.0f;
    }
}

// out[i,:] = log_softmax(relu(h2[i,:] + b))  — streaming logsumexp, no spills
__global__ void finalize_kernel(const float* __restrict__ h2,
                                const float* __restrict__ b,
                                float* __restrict__ out, int N, int C) {
    int i = blockIdx.x * blockDim.x + threadIdx.x;
    if (i >= N) return;
    const float* r = h2 + (size_t)i * C;
    float m = -3.402823466e38f, s = 0.0f;
    for (int c = 0; c < C; ++c) {
        float v = r[c] + b[c];
        v = v > 0.0f ? v : 0.0f;
        if (v > m) { s = s * expf(m - v) + 1.0f; m = v; }
        else       { s += expf(v - m); }
    }
    const float ls = logf(s);
    float* o = out + (size_t)i * C;
    for (int c = 0; c < C; ++c) {
        float v = r[c] + b[c];
        v = v > 0.0f ? v : 0.0f;
        o[c] = v - m - ls;
    }
}

// ---------------------------------------------------------------------------
static inline int cdiv(long long a, long long b) { return (int)((a + b - 1) / b); }

extern "C" void kernel_launch(void* const* d_in, const int* in_sizes, int n_in,
                              void* d_out, int out_size, void* d_ws, size_t ws_size,
                              hipStream_t stream) {
    const float* x  = (const float*)d_in[0];
    const int*   ei = (const int*)d_in[1];
    const float* W1 = (const float*)d_in[2];
    const float* b1 = (const float*)d_in[3];
    const float* W2 = (const float*)d_in[4];
    const float* b2 = (const float*)d_in[5];
    float* out = (float*)d_out;

    const int H   = in_sizes[3];            // 128
    const int C   = in_sizes[5];            // 40
    const int Fin = in_sizes[2] / H;        // 128
    const int N   = in_sizes[0] / Fin;      // 50000
    const int E   = in_sizes[1] / 2;        // 800000
    const int* row = ei;
    const int* col = ei + E;

    // workspace carve-out (256B aligned)
    char* ws = (char*)d_ws;
    size_t off = 0;
    auto carve = [&](size_t nfloats) {
        float* p = (float*)(ws + off);
        off += ((nfloats * sizeof(float) + 255) / 256) * 256;
        return p;
    };
    float* dis = carve((size_t)N);
    float* xw  = carve((size_t)N * H);
    float* h   = carve((size_t)N * H);
    float* hw  = carve((size_t)N * C);
    float* h2  = carve((size_t)N * C);
    (void)ws_size; (void)n_in; (void)out_size;

    const int B = 256;

    // normalization coefficients
    deg_init_kernel<<<cdiv(N, B), B, 0, stream>>>(dis, N);
    deg_edges_kernel<<<cdiv(E, B), B, 0, stream>>>(col, dis, E);
    rsqrt_kernel<<<cdiv(N, B), B, 0, stream>>>(dis, N);

    // ---- layer 1: h = relu(Aggr(x @ W1) + b1) ----
    {
        constexpr int NT = 8;  // H = 128 -> 8 n-tiles
        const size_t smem = (size_t)64 * (NT * 16 + 8) * sizeof(float);
        wmma_gemm_kernel<NT><<<cdiv(N, 128), B, smem, stream>>>(x, W1, xw, N, H, Fin);
    }
    selfloop_init_kernel<<<cdiv((long long)N * H, B), B, 0, stream>>>(xw, dis, h, N * H, H);
    edge_scatter_kernel<<<cdiv((long long)E * (H / 4), B), B, 0, stream>>>(
        xw, dis, row, col, h, E, H / 4, H);
    bias_relu_kernel<<<cdiv((long long)N * H, B), B, 0, stream>>>(h, b1, N * H, H);

    // ---- layer 2: out = log_softmax(relu(Aggr(h @ W2) + b2)) ----
    {
        constexpr int NT = 3;  // C = 40 -> 3 n-tiles (padded to 48)
        const size_t smem = (size_t)64 * (NT * 16 + 8) * sizeof(float);
        wmma_gemm_kernel<NT><<<cdiv(N, 128), B, smem, stream>>>(h, W2, hw, N, C, H);
    }
    selfloop_init_kernel<<<cdiv((long long)N * C, B), B, 0, stream>>>(hw, dis, h2, N * C, C);
    edge_scatter_kernel<<<cdiv((long long)E * (C / 4), B), B, 0, stream>>>(
        hw, dis, row, col, h2, E, C / 4, C);
    finalize_kernel<<<cdiv(N, B), B, 0, stream>>>(h2, b2, out, N, C);
}